// SAGEConv_6545530159133
// MI455X (gfx1250) — compile-verified
//
#include <hip/hip_runtime.h>

typedef float v2f __attribute__((ext_vector_type(2)));
typedef float v8f __attribute__((ext_vector_type(8)));

#define DF 128          // feature dim (D_IN == D_OUT == 128)
#define SCAN_BLK 256    // elements per scan1 block (must be 256: fixup uses i>>8)

// ---------------- zero an int array ----------------
__global__ void sage_zero_i(int* __restrict__ p, int n) {
    int i = blockIdx.x * blockDim.x + threadIdx.x;
    if (i < n) p[i] = 0;
}

// ---------------- histogram of destination rows ----------------
__global__ void sage_hist(const int* __restrict__ erow, int* __restrict__ counts, int E) {
    int e = blockIdx.x * blockDim.x + threadIdx.x;
    if (e < E) atomicAdd(counts + erow[e], 1);
}

// ---------------- scan pass 1: per-block inclusive scan of counts ----------------
// writes inclusive partial into rowptr[i+1]; block total into bsums[b]
__global__ void sage_scan1(const int* __restrict__ counts, int* __restrict__ rowptr,
                           int* __restrict__ bsums, int N) {
    __shared__ int s[SCAN_BLK];
    const int t = threadIdx.x;
    const int i = blockIdx.x * SCAN_BLK + t;
    int v = (i < N) ? counts[i] : 0;
    s[t] = v;
    __syncthreads();
    #pragma unroll
    for (int off = 1; off < SCAN_BLK; off <<= 1) {
        int u = (t >= off) ? s[t - off] : 0;
        __syncthreads();
        s[t] += u;
        __syncthreads();
    }
    if (i < N) rowptr[i + 1] = s[t];
    if (t == SCAN_BLK - 1) bsums[blockIdx.x] = s[SCAN_BLK - 1];
}

// ---------------- scan pass 2: single block scans block sums (nb <= 1024) -------
__global__ void sage_scan2(const int* __restrict__ bsums, int* __restrict__ boff, int nb) {
    __shared__ int s[1024];
    const int t = threadIdx.x;
    int v = (t < nb) ? bsums[t] : 0;
    s[t] = v;
    __syncthreads();
    #pragma unroll
    for (int off = 1; off < 1024; off <<= 1) {
        int u = (t >= off) ? s[t - off] : 0;
        __syncthreads();
        s[t] += u;
        __syncthreads();
    }
    if (t < nb) boff[t] = s[t] - v;   // exclusive block offset
}

// ---------------- scan pass 3: fixup + init placement cursors ----------------
__global__ void sage_fixup(int* __restrict__ rowptr, int* __restrict__ cursor,
                           const int* __restrict__ boff, int N) {
    int i = blockIdx.x * blockDim.x + threadIdx.x;
    if (i < N) {
        int v = rowptr[i + 1] + boff[i >> 8];
        rowptr[i + 1] = v;
        if (i + 1 < N) cursor[i + 1] = v;
        if (i == 0) { rowptr[0] = 0; cursor[0] = 0; }
    }
}

// ---------------- CSR placement ----------------
__global__ void sage_place(const int* __restrict__ erow, const int* __restrict__ ecol,
                           const float* __restrict__ ev, int* __restrict__ cursor,
                           int* __restrict__ ccol, float* __restrict__ cval, int E) {
    int e = blockIdx.x * blockDim.x + threadIdx.x;
    if (e < E) {
        int r = erow[e];
        int p = atomicAdd(cursor + r, 1);
        ccol[p] = ecol[e];
        cval[p] = ev[e];
    }
}

// ---------------- fused: gather-SpMM into LDS + dual WMMA GEMM + bias ----------
// Block = 256 threads = 8 waves, one 16-row output stripe.
// Phase 1: stage h[r0:r0+16, :] into LDS; compute neighbor rows (CSR gather,
//          wave wv owns local rows 2wv,2wv+1; lane = features [4l,4l+4)).
// Phase 2: wave wv computes out[r0:r0+16, 16wv:16wv+16] with
//          V_WMMA_F32_16X16X4_F32 (A from LDS, B = W rows, fp32 accumulate).
__global__ void sage_fused(const float* __restrict__ h,
                           const int*   __restrict__ rowptr,
                           const int*   __restrict__ ccol,
                           const float* __restrict__ cval,
                           const float* __restrict__ Wself,
                           const float* __restrict__ bself,
                           const float* __restrict__ Wnb,
                           const float* __restrict__ bnb,
                           float* __restrict__ out, int N) {
    __shared__ float sh[16 * DF];   // h tile        (8 KB)
    __shared__ float sn[16 * DF];   // neighbor tile (8 KB)

    const int r0   = blockIdx.x * 16;
    const int lane = threadIdx.x & 31;
    const int wv   = threadIdx.x >> 5;      // 0..7
    const bool full = (r0 + 16 <= N);

    // stage h tile (float4 -> global_load_b128 / ds_store_b128)
    for (int i = threadIdx.x; i < (16 * DF) / 4; i += blockDim.x) {
        int r  = (i * 4) / DF;
        int cc = (i * 4) % DF;
        int grow = r0 + r;
        float4 a = make_float4(0.f, 0.f, 0.f, 0.f);
        if (full || grow < N) a = *(const float4*)(h + (size_t)grow * DF + cc);
        ((float4*)sh)[i] = a;
    }

    // neighbor rows via CSR gather (deterministic per-row order, no f32 atomics)
    #pragma unroll
    for (int rr = 0; rr < 2; ++rr) {
        const int rl   = 2 * wv + rr;
        const int grow = r0 + rl;
        float4 acc = make_float4(0.f, 0.f, 0.f, 0.f);
        if (grow < N) {
            const int e1 = rowptr[grow + 1];
            for (int e = rowptr[grow]; e < e1; ++e) {
                int   c = ccol[e];          // wave-uniform
                float v = cval[e];          // wave-uniform
                float4 m = ((const float4*)(h + (size_t)c * DF))[lane];
                acc.x += v * m.x; acc.y += v * m.y;
                acc.z += v * m.z; acc.w += v * m.w;
            }
        }
        *(float4*)(sn + rl * DF + lane * 4) = acc;
    }
    __syncthreads();

    const int c0   = wv * 16;            // output column tile
    const int mrow = lane & 15;          // M for A, N for B
    const int koff = (lane >> 4) * 2;    // K sub-stripe per lane half

    v8f acc = {};
    #pragma unroll 4
    for (int k0 = 0; k0 < DF; k0 += 4) {
        const int kk = k0 + koff;
        v2f a  = *(const v2f*)(sh + mrow * DF + kk);
        v2f b  = *(const v2f*)(Wself + (size_t)(c0 + mrow) * DF + kk);
        acc = __builtin_amdgcn_wmma_f32_16x16x4_f32(false, a, false, b,
                                                    (short)0, acc, false, false);
        v2f a2 = *(const v2f*)(sn + mrow * DF + kk);
        v2f b2 = *(const v2f*)(Wnb + (size_t)(c0 + mrow) * DF + kk);
        acc = __builtin_amdgcn_wmma_f32_16x16x4_f32(false, a2, false, b2,
                                                    (short)0, acc, false, false);
    }

    const float bias   = bself[c0 + mrow] + bnb[c0 + mrow];
    const int   rowadd = (lane >> 4) * 8;
    if (full) {
        #pragma unroll
        for (int r = 0; r < 8; ++r)
            out[(size_t)(r0 + r + rowadd) * DF + c0 + mrow] = acc[r] + bias;
    } else {
        #pragma unroll
        for (int r = 0; r < 8; ++r) {
            int grow = r0 + r + rowadd;
            if (grow < N) out[(size_t)grow * DF + c0 + mrow] = acc[r] + bias;
        }
    }
}

extern "C" void kernel_launch(void* const* d_in, const int* in_sizes, int n_in,
                              void* d_out, int out_size, void* d_ws, size_t ws_size,
                              hipStream_t stream) {
    const float* h     = (const float*)d_in[0];
    const float* ev    = (const float*)d_in[1];
    const float* Wself = (const float*)d_in[2];
    const float* bself = (const float*)d_in[3];
    const float* Wnb   = (const float*)d_in[4];
    const float* bnb   = (const float*)d_in[5];
    const int*   erow  = (const int*)d_in[6];
    const int*   ecol  = (const int*)d_in[7];
    float* out = (float*)d_out;

    const int D = in_sizes[3];         // 128
    const int N = in_sizes[0] / D;     // 100000
    const int E = in_sizes[1];         // 1600000

    // ---- workspace carve-out (256B aligned slices) ----
    char* wsp = (char*)d_ws;
    auto carve = [&](size_t bytes) -> void* {
        void* p = wsp;
        wsp += (bytes + 255) & ~(size_t)255;
        return p;
    };
    const int nb1 = (N + SCAN_BLK - 1) / SCAN_BLK;       // 391 (<= 1024 required)
    int*   counts = (int*)carve((size_t)N * 4);
    int*   rowptr = (int*)carve((size_t)(N + 1) * 4);
    int*   cursor = (int*)carve((size_t)N * 4);
    int*   bsums  = (int*)carve((size_t)nb1 * 4);
    int*   boff   = (int*)carve((size_t)nb1 * 4);
    int*   ccol   = (int*)carve((size_t)E * 4);
    float* cval   = (float*)carve((size_t)E * 4);

    // ---- CSR build ----
    sage_zero_i<<<(N + 255) / 256, 256, 0, stream>>>(counts, N);
    sage_hist  <<<(E + 255) / 256, 256, 0, stream>>>(erow, counts, E);
    sage_scan1 <<<nb1, SCAN_BLK, 0, stream>>>(counts, rowptr, bsums, N);
    sage_scan2 <<<1, 1024, 0, stream>>>(bsums, boff, nb1);
    sage_fixup <<<(N + 255) / 256, 256, 0, stream>>>(rowptr, cursor, boff, N);
    sage_place <<<(E + 255) / 256, 256, 0, stream>>>(erow, ecol, ev, cursor, ccol, cval, E);

    // ---- fused gather-SpMM + WMMA GEMM ----
    sage_fused <<<(N + 15) / 16, 256, 0, stream>>>(h, rowptr, ccol, cval,
                                                   Wself, bself, Wnb, bnb, out, N);
}